// BitNetSummaryEncoder_43576738185550
// MI455X (gfx1250) — compile-verified
//
#include <hip/hip_runtime.h>
#include <math.h>

typedef float v2f __attribute__((ext_vector_type(2)));
typedef float v4f __attribute__((ext_vector_type(4)));
typedef float v8f __attribute__((ext_vector_type(8)));

// ---------------- workspace layout (floats) ----------------
#define N_WFRAG   (32 * 13 * 32 * 2)   // 26624 : B-fragments, frag order
#define OFF_WFRAG 0
#define OFF_QVW1  (N_WFRAG)            // 8x4
#define OFF_QVW2  (OFF_QVW1 + 32)      // 6x8
#define OFF_QPW1  (OFF_QVW2 + 48)      // 24x12
#define OFF_QPW2  (OFF_QPW1 + 288)     // 20x24

#define ZS 260                         // z-stage row stride (floats): 8*260 % 64 == 32 -> bank-split halves

__device__ __forceinline__ float gelu_exact(float x) {
    return 0.5f * x * (1.0f + erff(x * 0.70710678118654752440f));
}

__device__ __forceinline__ float fast_sigmoid(float x) {
    // v_exp_f32 + v_rcp_f32 path (~1 ulp each)
    return __builtin_amdgcn_rcpf(1.0f + __expf(-x));
}

// ============================================================
// Prep: ternary-quantize small MLP weights + build WMMA B-fragments
// frag index: ((n*13 + k)*32 + lane)*2 + r
//   K = 4k + (lane<16 ? r : 2+r), col = 16n + (lane&15)
//   value = col<256 ? f_wh[col][K] : f_wg[col-256][K], 0 for K==51
// ============================================================
__global__ void bitnet_prep_kernel(const float* __restrict__ vw1,
                                   const float* __restrict__ vw2,
                                   const float* __restrict__ pw1,
                                   const float* __restrict__ pw2,
                                   const float* __restrict__ fwh,
                                   const float* __restrict__ fwg,
                                   float* __restrict__ ws) {
    // ternary quantization (4 tiny tensors, one thread each)
    if (blockIdx.x == 0 && threadIdx.x < 4) {
        const float* src; float* dst; int n;
        switch (threadIdx.x) {
            case 0:  src = vw1; dst = ws + OFF_QVW1; n = 32;  break;
            case 1:  src = vw2; dst = ws + OFF_QVW2; n = 48;  break;
            case 2:  src = pw1; dst = ws + OFF_QPW1; n = 288; break;
            default: src = pw2; dst = ws + OFF_QPW2; n = 480; break;
        }
        float acc = 0.f;
        for (int i = 0; i < n; ++i) acc += fabsf(src[i]);
        float scale = acc / (float)n;
        float inv = 1.0f / (scale + 1e-5f);
        for (int i = 0; i < n; ++i) {
            float q = rintf(src[i] * inv);          // round-half-even, like jnp.round
            q = fminf(1.0f, fmaxf(-1.0f, q));
            dst[i] = q * scale;
        }
    }
    // B-fragment layout build
    int idx = blockIdx.x * blockDim.x + threadIdx.x;
    if (idx < N_WFRAG) {
        int r = idx & 1;
        int t = idx >> 1;
        int lane = t & 31; t >>= 5;
        int k = t % 13;
        int n = t / 13;
        int K = 4 * k + ((lane < 16) ? r : 2 + r);
        int col = 16 * n + (lane & 15);
        float val = 0.0f;
        if (K < 51)
            val = (col < 256) ? fwh[col * 51 + K] : fwg[(col - 256) * 51 + K];
        ws[OFF_WFRAG + idx] = val;
    }
}

// ============================================================
// Main: 64 threads = 2 waves, each wave owns a 16-row tile.
// ============================================================
__global__ __launch_bounds__(64)
void bitnet_encoder_kernel(const int* __restrict__ rdc, const int* __restrict__ wrc,
                           const int* __restrict__ ftc, const int* __restrict__ cwc,
                           const int* __restrict__ rcy,
                           const float* __restrict__ vol, const float* __restrict__ pres,
                           const float* __restrict__ cemb, const float* __restrict__ remb,
                           const float* __restrict__ p_b1, const float* __restrict__ p_b2,
                           const float* __restrict__ v_b1, const float* __restrict__ v_b2,
                           const float* __restrict__ f_bh, const float* __restrict__ f_bg,
                           const float* __restrict__ ln_g, const float* __restrict__ ln_b,
                           const float* __restrict__ ws, float* __restrict__ out, int B) {
    __shared__ float sh_comb[2][16][52];
    __shared__ float sh_z[2][16 * ZS];
    __shared__ float sh_stats[2][16][2];

    const int lane = threadIdx.x & 31;
    const int wave = threadIdx.x >> 5;
    const int half = lane >> 4;
    const int l16  = lane & 15;
    const long tile    = (long)blockIdx.x * 2 + wave;
    const long rowbase = tile * 16;

    const float* qvw1 = ws + OFF_QVW1;
    const float* qvw2 = ws + OFF_QVW2;
    const float* qpw1 = ws + OFF_QPW1;
    const float* qpw2 = ws + OFF_QPW2;

    // ---------- phase 1: per-row features, split across lane halves ----------
    // half 0 (lanes 0..15):  embeddings + volatility MLP -> comb[0..30], comb[51]
    // half 1 (lanes 16..31): pressure MLP               -> comb[31..50]
    {
        long row = rowbase + l16;
        if (row >= B) row = B - 1;
        if (half == 0) {
            int i0 = rdc[row], i1 = wrc[row], i2 = ftc[row], i3 = cwc[row], i4 = rcy[row];
            #pragma unroll
            for (int i = 0; i < 5; ++i) {
                sh_comb[wave][l16][i]      = cemb[i0 * 5 + i];
                sh_comb[wave][l16][5 + i]  = cemb[i1 * 5 + i];
                sh_comb[wave][l16][10 + i] = cemb[i2 * 5 + i];
                sh_comb[wave][l16][15 + i] = cemb[i3 * 5 + i];
                sh_comb[wave][l16][20 + i] = remb[i4 * 5 + i];
            }
            // volatility MLP: 4 -> 8 (gelu) -> 6
            float x4[4];
            #pragma unroll
            for (int i = 0; i < 4; ++i) x4[i] = vol[row * 4 + i];
            float h8[8];
            #pragma unroll
            for (int o = 0; o < 8; ++o) {
                float s = v_b1[o];
                #pragma unroll
                for (int i = 0; i < 4; ++i) s += qvw1[o * 4 + i] * x4[i];
                h8[o] = gelu_exact(s);
            }
            #pragma unroll
            for (int o = 0; o < 6; ++o) {
                float s = v_b2[o];
                #pragma unroll
                for (int i = 0; i < 8; ++i) s += qvw2[o * 8 + i] * h8[i];
                sh_comb[wave][l16][25 + o] = s;
            }
            sh_comb[wave][l16][51] = 0.0f;
        } else {
            // pressure MLP: 12 -> 24 (gelu) -> 20
            float x12[12];
            #pragma unroll
            for (int i = 0; i < 12; ++i) x12[i] = pres[row * 12 + i];
            float h24[24];
            #pragma unroll
            for (int o = 0; o < 24; ++o) {
                float s = p_b1[o];
                #pragma unroll
                for (int i = 0; i < 12; ++i) s += qpw1[o * 12 + i] * x12[i];
                h24[o] = gelu_exact(s);
            }
            #pragma unroll
            for (int o = 0; o < 20; ++o) {
                float s = p_b2[o];
                #pragma unroll
                for (int i = 0; i < 24; ++i) s += qpw2[o * 24 + i] * h24[i];
                sh_comb[wave][l16][31 + o] = s;
            }
        }
    }
    __syncthreads();

    // ---------- phase 2: A fragments (16x4 f32 tiles) kept resident ----------
    // lanes 0-15: M=lane, VGPR0=K(4k), VGPR1=K(4k+1); lanes 16-31: M=lane-16, K(4k+2),K(4k+3)
    v2f afrag[13];
    #pragma unroll
    for (int k = 0; k < 13; ++k)
        afrag[k] = *(const v2f*)&sh_comb[wave][l16][4 * k + 2 * half];

    const v2f* wfrag = (const v2f*)(ws + OFF_WFRAG);
    float sum[8], sq[8], mu8[8], rs8[8];
    #pragma unroll
    for (int j = 0; j < 8; ++j) { sum[j] = 0.f; sq[j] = 0.f; }

    // ---------- phase 3: 16 column tiles; h-tile (n) and gate-tile (n+16) together ----------
    for (int n = 0; n < 16; ++n) {
        v8f acch = {};
        v8f accg = {};
        const v2f* bh = wfrag + (size_t)n * 13 * 32 + lane;
        const v2f* bg = wfrag + (size_t)(n + 16) * 13 * 32 + lane;
        #pragma unroll
        for (int k = 0; k < 13; ++k) {
            v2f vbh = bh[k * 32];
            v2f vbg = bg[k * 32];
            acch = __builtin_amdgcn_wmma_f32_16x16x4_f32(false, afrag[k], false, vbh,
                                                         (short)0, acch, false, false);
            accg = __builtin_amdgcn_wmma_f32_16x16x4_f32(false, afrag[k], false, vbg,
                                                         (short)0, accg, false, false);
        }
        const int col = 16 * n + l16;
        const float bch = f_bh[col];
        const float bcg = f_bg[col];
        #pragma unroll
        for (int j = 0; j < 8; ++j) {
            float h  = acch[j] + bch;
            float gl = accg[j] + bcg;
            float g  = fast_sigmoid(gl);
            float z  = g * h;
            sum[j] += z;
            sq[j]  += z * z;
            sh_z[wave][(j + 8 * half) * ZS + col] = z;   // D layout: rows j / j+8 per half
        }
    }

    // ---------- phase 4: LayerNorm stats (reduce across 16 lanes of each half) ----------
    #pragma unroll
    for (int j = 0; j < 8; ++j) {
        float s = sum[j], q = sq[j];
        #pragma unroll
        for (int m = 8; m >= 1; m >>= 1) {
            s += __shfl_xor(s, m, 32);
            q += __shfl_xor(q, m, 32);
        }
        float mu  = s * (1.0f / 256.0f);
        float var = q * (1.0f / 256.0f) - mu * mu;
        mu8[j] = mu;
        rs8[j] = rsqrtf(var + 1e-5f);
    }
    if (l16 == 0) {
        #pragma unroll
        for (int j = 0; j < 8; ++j) {
            sh_stats[wave][j + 8 * half][0] = mu8[j];
            sh_stats[wave][j + 8 * half][1] = rs8[j];
        }
    }
    __syncthreads();

    // ---------- phase 5: normalize + coalesced 512B/row stores ----------
    #pragma unroll
    for (int cc = 0; cc < 2; ++cc) {
        const int col = cc * 128 + lane * 4;
        v4f lg = *(const v4f*)&ln_g[col];
        v4f lb = *(const v4f*)&ln_b[col];
        #pragma unroll
        for (int r = 0; r < 16; ++r) {
            long row = rowbase + r;
            if (row < B) {
                float mu   = sh_stats[wave][r][0];
                float rstd = sh_stats[wave][r][1];
                v4f z = *(const v4f*)&sh_z[wave][r * ZS + col];
                v4f o = (z - mu) * rstd * lg + lb;
                *(v4f*)&out[(size_t)row * 256 + col] = o;
            }
        }
    }
}

// ============================================================
extern "C" void kernel_launch(void* const* d_in, const int* in_sizes, int n_in,
                              void* d_out, int out_size, void* d_ws, size_t ws_size,
                              hipStream_t stream) {
    const int*   rdc  = (const int*)d_in[0];
    const int*   wrc  = (const int*)d_in[1];
    const int*   ftc  = (const int*)d_in[2];
    const int*   cwc  = (const int*)d_in[3];
    const int*   rcy  = (const int*)d_in[4];
    const float* vol  = (const float*)d_in[5];
    const float* pres = (const float*)d_in[6];
    const float* cemb = (const float*)d_in[7];
    const float* remb = (const float*)d_in[8];
    const float* pw1  = (const float*)d_in[9];
    const float* pb1  = (const float*)d_in[10];
    const float* pw2  = (const float*)d_in[11];
    const float* pb2  = (const float*)d_in[12];
    const float* vw1  = (const float*)d_in[13];
    const float* vb1  = (const float*)d_in[14];
    const float* vw2  = (const float*)d_in[15];
    const float* vb2  = (const float*)d_in[16];
    const float* fwh  = (const float*)d_in[17];
    const float* fbh  = (const float*)d_in[18];
    const float* fwg  = (const float*)d_in[19];
    const float* fbg  = (const float*)d_in[20];
    const float* lng  = (const float*)d_in[21];
    const float* lnb  = (const float*)d_in[22];

    const int B = in_sizes[0];
    float* ws = (float*)d_ws;

    const int prep_blocks = (N_WFRAG + 255) / 256;
    bitnet_prep_kernel<<<prep_blocks, 256, 0, stream>>>(vw1, vw2, pw1, pw2, fwh, fwg, ws);

    const int main_blocks = (B + 31) / 32;   // 32 rows per block (2 waves x 16)
    bitnet_encoder_kernel<<<main_blocks, 64, 0, stream>>>(
        rdc, wrc, ftc, cwc, rcy, vol, pres, cemb, remb,
        pb1, pb2, vb1, vb2, fbh, fbg, lng, lnb,
        ws, (float*)d_out, B);
}